// LocalMultiRangeAttention_53764400611379
// MI455X (gfx1250) — compile-verified
//
#include <hip/hip_runtime.h>
#include <hip/hip_bf16.h>
#include <math.h>

typedef float v2f __attribute__((ext_vector_type(2)));
typedef float v8f __attribute__((ext_vector_type(8)));

#define DIMC   256
#define HDIM   32
#define NHEADS 8
#define HW     3136
#define WIDTH  56
#define SCALE  0.17677669529663687f   // 32^-0.5

// ---------------------------------------------------------------------------
// Wave-level MTx16 (M) x NTx16 (N) fp32 GEMM macro-tile over K using
// V_WMMA_F32_16X16X4_F32 with register blocking: MT A-operands and NT
// B-operands are loaded once per K-step and shared across MT*NT wmmas.
// A: row-major M x K (lda). B: row-major K x N (ldb).
// Per ISA VGPR layout: lane = (idx = lane&15, khalf = lane>>4)
//   A regs = A[idx][2*khalf + {0,1}]   (contiguous pair -> v2f load)
//   B regs = B[2*khalf + {0,1}][idx]   (two strided b32 loads)
// ---------------------------------------------------------------------------
template<int MT, int NT>
__device__ __forceinline__ void wmma_gemm_multi(const float* __restrict__ A, int lda,
                                                const float* __restrict__ B, int ldb,
                                                int K, v8f acc[MT][NT])
{
    const int lane = threadIdx.x & 31;
    const int idx  = lane & 15;
    const int kb   = (lane >> 4) << 1;
    const float* ap = A + (size_t)idx * lda + kb;
    const float* bp = B + (size_t)kb * ldb + idx;
#pragma unroll 4
    for (int k = 0; k < K; k += 4) {
        v2f a[MT];
#pragma unroll
        for (int mi = 0; mi < MT; ++mi)
            a[mi] = *(const v2f*)(ap + (size_t)mi * 16 * lda);
        v2f bv[NT];
#pragma unroll
        for (int ni = 0; ni < NT; ++ni) {
            bv[ni].x = bp[ni * 16];
            bv[ni].y = bp[ni * 16 + ldb];
        }
#pragma unroll
        for (int mi = 0; mi < MT; ++mi)
#pragma unroll
            for (int ni = 0; ni < NT; ++ni)
                acc[mi][ni] = __builtin_amdgcn_wmma_f32_16x16x4_f32(
                    /*neg_a=*/false, a[mi], /*neg_b=*/false, bv[ni],
                    /*c_mod=*/(short)0, acc[mi][ni],
                    /*reuse_a=*/false, /*reuse_b=*/false);
        ap += 4;
        bp += (size_t)4 * ldb;
    }
}

// ---------------------------------------------------------------------------
// Kernel A: qkv = w_qkv @ x + b_qkv    (M=768, K=256, N=3136 per batch)
// Wave macro-tile: 32(M) x 64(N).  768 = 24 mBlocks, 3136 = 49 nQuads.
// Output layout: qkvbuf[b][which(3)][head(8)][p(3136)][d(32)]  (d contiguous)
// grid = (49, 6, 4), block = 128 (4 waves; wave -> mBlock)
// ---------------------------------------------------------------------------
__global__ void __launch_bounds__(128)
qkv_kernel(const float* __restrict__ x,      // (B,256,56,56)
           const float* __restrict__ w_qkv,  // (768,256)
           const float* __restrict__ b_qkv,  // (768)
           float* __restrict__ qkvbuf)
{
    const int wave   = threadIdx.x >> 5;
    const int lane   = threadIdx.x & 31;
    const int nQuad  = blockIdx.x;                 // 0..48
    const int mBlock = blockIdx.y * 4 + wave;      // 0..23
    const int b      = blockIdx.z;

    const float* A = w_qkv + (size_t)mBlock * 32 * DIMC;
    const float* B = x + (size_t)b * DIMC * HW + nQuad * 64;

    v8f acc[2][4] = {};
    wmma_gemm_multi<2, 4>(A, DIMC, B, HW, DIMC, acc);

    const int idx   = lane & 15;
    const int khalf = lane >> 4;

#pragma unroll
    for (int mi = 0; mi < 2; ++mi) {
        const int m0    = mBlock * 32 + mi * 16 + khalf * 8;  // multiple of 8
        const int which = m0 >> 8;                            // 0=q 1=k 2=v
        const int head  = (m0 >> 5) & 7;
        const int d0    = m0 & 31;                            // 0,8,16,24
        float4 bias_lo, bias_hi;
        bias_lo.x = b_qkv[m0 + 0]; bias_lo.y = b_qkv[m0 + 1];
        bias_lo.z = b_qkv[m0 + 2]; bias_lo.w = b_qkv[m0 + 3];
        bias_hi.x = b_qkv[m0 + 4]; bias_hi.y = b_qkv[m0 + 5];
        bias_hi.z = b_qkv[m0 + 6]; bias_hi.w = b_qkv[m0 + 7];
#pragma unroll
        for (int ni = 0; ni < 4; ++ni) {
            const int p = nQuad * 64 + ni * 16 + idx;
            float* outp = qkvbuf +
                ((((size_t)(b * 3 + which) * NHEADS + head) * HW + p) * HDIM + d0);
            float4 lo, hi;
            lo.x = acc[mi][ni][0] + bias_lo.x;
            lo.y = acc[mi][ni][1] + bias_lo.y;
            lo.z = acc[mi][ni][2] + bias_lo.z;
            lo.w = acc[mi][ni][3] + bias_lo.w;
            hi.x = acc[mi][ni][4] + bias_hi.x;
            hi.y = acc[mi][ni][5] + bias_hi.y;
            hi.z = acc[mi][ni][6] + bias_hi.z;
            hi.w = acc[mi][ni][7] + bias_hi.w;
            *(float4*)(outp)     = lo;
            *(float4*)(outp + 4) = hi;
        }
    }
}

// ---------------------------------------------------------------------------
// Kernel B: neighborhood attention, one thread per (b, head, pixel).
// Zero-padded neighbors (as in the reference) contribute logit=0 and v=0 to
// the softmax. Single-pass online softmax, q + 32 accumulators in registers.
// Output written channel-major: attnbuf[b][c=head*32+d][p].
// ---------------------------------------------------------------------------
__global__ void __launch_bounds__(256)
attn_kernel(const float* __restrict__ qkvbuf,
            float* __restrict__ attnbuf)
{
    const int tid  = blockIdx.x * blockDim.x + threadIdx.x;  // < 4*8*3136
    const int p    = tid % HW;
    const int bh   = tid / HW;
    const int head = bh & 7;
    const int b    = bh >> 3;
    const int yy0  = p / WIDTH;
    const int xx0  = p % WIDTH;

    const int ksz = (head < 4) ? 5 : ((head < 7) ? 7 : 9);
    const int pad = ksz >> 1;

    const float* qb = qkvbuf + (((size_t)(b * 3 + 0) * NHEADS + head) * HW + p) * HDIM;
    const float* kb = qkvbuf + (((size_t)(b * 3 + 1) * NHEADS + head) * HW) * HDIM;
    const float* vb = qkvbuf + (((size_t)(b * 3 + 2) * NHEADS + head) * HW) * HDIM;

    float q[HDIM];
#pragma unroll
    for (int d4 = 0; d4 < 8; ++d4) {
        float4 t = ((const float4*)qb)[d4];
        q[4 * d4 + 0] = t.x * SCALE;
        q[4 * d4 + 1] = t.y * SCALE;
        q[4 * d4 + 2] = t.z * SCALE;
        q[4 * d4 + 3] = t.w * SCALE;
    }

    float mx   = -INFINITY;
    float ssum = 0.0f;
    float acc[HDIM];
#pragma unroll
    for (int d = 0; d < HDIM; ++d) acc[d] = 0.0f;

    for (int ki = 0; ki < ksz; ++ki) {
        const int yy = yy0 + ki - pad;
        for (int kj = 0; kj < ksz; ++kj) {
            const int xx = xx0 + kj - pad;
            const bool inb = (yy >= 0) && (yy < WIDTH) && (xx >= 0) && (xx < WIDTH);
            const size_t np = (size_t)(inb ? (yy * WIDTH + xx) : 0) * HDIM;

            float s = 0.0f;
            if (inb) {
                const float* kp = kb + np;
#pragma unroll
                for (int d4 = 0; d4 < 8; ++d4) {
                    float4 t = ((const float4*)kp)[d4];
                    s += q[4 * d4 + 0] * t.x + q[4 * d4 + 1] * t.y
                       + q[4 * d4 + 2] * t.z + q[4 * d4 + 3] * t.w;
                }
            }
            // zero-padded OOB neighbor => s = 0, v = 0 (still in softmax!)
            const float nm = fmaxf(mx, s);
            const float cs = __expf(mx - nm);   // exp(-inf)=0 on first iter
            const float w  = __expf(s - nm);
            ssum = ssum * cs + w;
            if (inb) {
                const float* vp = vb + np;
#pragma unroll
                for (int d4 = 0; d4 < 8; ++d4) {
                    float4 t = ((const float4*)vp)[d4];
                    acc[4 * d4 + 0] = acc[4 * d4 + 0] * cs + w * t.x;
                    acc[4 * d4 + 1] = acc[4 * d4 + 1] * cs + w * t.y;
                    acc[4 * d4 + 2] = acc[4 * d4 + 2] * cs + w * t.z;
                    acc[4 * d4 + 3] = acc[4 * d4 + 3] * cs + w * t.w;
                }
            } else {
#pragma unroll
                for (int d = 0; d < HDIM; ++d) acc[d] *= cs;
            }
            mx = nm;
        }
    }

    const float inv = 1.0f / ssum;
    float* ob = attnbuf + ((size_t)b * DIMC + head * HDIM) * HW + p;
#pragma unroll
    for (int d = 0; d < HDIM; ++d)
        ob[(size_t)d * HW] = acc[d] * inv;
}

// ---------------------------------------------------------------------------
// Kernel C: out = w_proj @ attn + b_proj  (M=256, K=256, N=3136 per batch)
// Wave macro-tile: 32(M) x 64(N).  256 = 8 mBlocks, 3136 = 49 nQuads.
// Output: reference layout (B,256,56,56).
// grid = (49, 2, 4), block = 128 (4 waves)
// ---------------------------------------------------------------------------
__global__ void __launch_bounds__(128)
proj_kernel(const float* __restrict__ attnbuf, // (B,256,3136) channel-major
            const float* __restrict__ w_proj,  // (256,256)
            const float* __restrict__ b_proj,  // (256)
            float* __restrict__ out)
{
    const int wave   = threadIdx.x >> 5;
    const int lane   = threadIdx.x & 31;
    const int nQuad  = blockIdx.x;                 // 0..48
    const int mBlock = blockIdx.y * 4 + wave;      // 0..7
    const int b      = blockIdx.z;

    const float* A = w_proj + (size_t)mBlock * 32 * DIMC;
    const float* B = attnbuf + (size_t)b * DIMC * HW + nQuad * 64;

    v8f acc[2][4] = {};
    wmma_gemm_multi<2, 4>(A, DIMC, B, HW, DIMC, acc);

    const int idx   = lane & 15;
    const int khalf = lane >> 4;

#pragma unroll
    for (int mi = 0; mi < 2; ++mi) {
        const int m0 = mBlock * 32 + mi * 16 + khalf * 8;
#pragma unroll
        for (int ni = 0; ni < 4; ++ni) {
            const int p = nQuad * 64 + ni * 16 + idx;
#pragma unroll
            for (int r = 0; r < 8; ++r)
                out[((size_t)b * DIMC + m0 + r) * HW + p] = acc[mi][ni][r] + b_proj[m0 + r];
        }
    }
}

// ---------------------------------------------------------------------------
extern "C" void kernel_launch(void* const* d_in, const int* in_sizes, int n_in,
                              void* d_out, int out_size, void* d_ws, size_t ws_size,
                              hipStream_t stream)
{
    const float* x      = (const float*)d_in[0];  // (4,256,56,56)
    const float* w_qkv  = (const float*)d_in[1];  // (768,256)
    const float* b_qkv  = (const float*)d_in[2];  // (768)
    const float* w_proj = (const float*)d_in[3];  // (256,256)
    const float* b_proj = (const float*)d_in[4];  // (256)
    float* out = (float*)d_out;

    float* qkvbuf  = (float*)d_ws;                                // 3*4*8*3136*32 floats
    float* attnbuf = qkvbuf + (size_t)3 * 4 * NHEADS * HW * HDIM; // +4*256*3136 floats

    // QKV projection: 24 mBlocks x 49 nQuads x 4 batches, 4 waves/block
    qkv_kernel<<<dim3(49, 6, 4), dim3(128), 0, stream>>>(x, w_qkv, b_qkv, qkvbuf);

    // Neighborhood attention: 4*8*3136 threads
    attn_kernel<<<dim3((4 * NHEADS * HW) / 256), dim3(256), 0, stream>>>(qkvbuf, attnbuf);

    // Output projection: 8 mBlocks x 49 nQuads x 4 batches
    proj_kernel<<<dim3(49, 2, 4), dim3(128), 0, stream>>>(attnbuf, w_proj, b_proj, out);
}